// GCNModel_3917010174092
// MI455X (gfx1250) — compile-verified
//
#include <hip/hip_runtime.h>

// ---------------------------------------------------------------------------
// GCN (2x GCNConv + FC) for MI455X / gfx1250.
// fp32 throughout; dense GEMMs use V_WMMA_F32_16X16X4_F32 (wave32 WMMA) with
// the shared A-tile staged in LDS via GLOBAL_LOAD_ASYNC_TO_LDS_B128;
// sparse aggregation uses global_atomic_add_f32 (L2-resident working set).
// ---------------------------------------------------------------------------

typedef float v2f __attribute__((ext_vector_type(2)));
typedef float v8f __attribute__((ext_vector_type(8)));

#define N_NODES 50000
#define N_EDGES 800000
#define DIM     128
#define FC_OUT  64
#define LEAKY   0.01f

#if defined(__HIP_DEVICE_COMPILE__) && __has_builtin(__builtin_amdgcn_global_load_async_to_lds_b128)
#define USE_ASYNC_LDS 1
// builtin signature (from hipcc diagnostic): (int4 AS1*, int4 AS3*, imm, imm)
typedef int v4i_gcc __attribute__((vector_size(16)));
typedef __attribute__((address_space(1))) v4i_gcc GV4;   // global int4
typedef __attribute__((address_space(3))) v4i_gcc LV4;   // LDS int4
#else
#define USE_ASYNC_LDS 0
#endif

__device__ __forceinline__ void wait_async_then_barrier() {
#if USE_ASYNC_LDS
#if __has_builtin(__builtin_amdgcn_s_wait_asynccnt)
    __builtin_amdgcn_s_wait_asynccnt(0);
#else
    asm volatile("s_wait_asynccnt 0" ::: "memory");
#endif
#endif
    __syncthreads();
}

// ---------------- degree / normalization ----------------

__global__ void init_deg_kernel(float* __restrict__ deg, int n) {
    int i = blockIdx.x * blockDim.x + threadIdx.x;
    if (i < n) deg[i] = 1.0f;                    // self-loop contributes 1
}

__global__ void deg_accum_kernel(const long long* __restrict__ dst,
                                 float* __restrict__ deg, int e) {
    int i = blockIdx.x * blockDim.x + threadIdx.x;
    if (i < e) atomicAdd(&deg[(int)dst[i]], 1.0f);
}

__global__ void rsqrt_kernel(float* __restrict__ deg, int n) {
    int i = blockIdx.x * blockDim.x + threadIdx.x;
    if (i < n) deg[i] = rsqrtf(deg[i]);          // deg >= 1 always (self loops)
}

// ---------------- WMMA GEMM: C[M,N] = A[M,K] @ B[K,N] (+ bias) ----------------
// One block = one 16-row strip of C; wave w owns column tile w.
// The 16xK A tile (shared by all waves) is staged once in LDS (async copy),
// then each wave runs K/4 V_WMMA_F32_16X16X4_F32.

template<int K, int N, bool HAS_BIAS>
__global__ void __launch_bounds__(32 * (N / 16))
gemm_wmma_f32(const float* __restrict__ A,
              const float* __restrict__ B,
              const float* __restrict__ bias,
              float* __restrict__ C) {
    constexpr int WAVES   = N / 16;
    constexpr int THREADS = 32 * WAVES;
    constexpr int PAD     = K + 4;               // 132: 4-bank row skew, conflict-free
    __shared__ __align__(16) float sA[16 * PAD];

    const int tm   = blockIdx.x;                 // row tile
    const int tn   = threadIdx.x >> 5;           // column tile = wave id
    const int lane = threadIdx.x & 31;

    // ---- cooperative A-tile stage: 16 rows x K cols, b128 per thread-chunk ----
    constexpr int F4_PER_ROW = K / 4;
    constexpr int CH = (16 * F4_PER_ROW) / THREADS;
#pragma unroll
    for (int c = 0; c < CH; ++c) {
        int idx = threadIdx.x + c * THREADS;     // float4 index in tile
        int r   = idx / F4_PER_ROW;
        int kk  = (idx % F4_PER_ROW) * 4;
        const float* gp = A + (size_t)(tm * 16 + r) * K + kk;
        float*       lp = &sA[r * PAD + kk];
#if USE_ASYNC_LDS
        __builtin_amdgcn_global_load_async_to_lds_b128((GV4*)gp, (LV4*)lp, 0, 0);
#else
        *(float4*)lp = *(const float4*)gp;
#endif
    }
    wait_async_then_barrier();

    // ---- WMMA main loop ----
    const int half = lane >> 4;                  // 0: lanes 0-15, 1: lanes 16-31
    const int l15  = lane & 15;
    const int col  = tn * 16 + l15;              // B/C column this lane owns

    // A frag (ISA 7.12.2): lanes 0-15 hold K=k0,k0+1 ; lanes 16-31 hold K=k0+2,k0+3
    const float* ar = &sA[l15 * PAD + half * 2];
    const float* bp = B + (size_t)(half * 2) * N + col;

    v8f acc = {};
#pragma unroll
    for (int k0 = 0; k0 < K; k0 += 4) {
        v2f a; a[0] = ar[k0]; a[1] = ar[k0 + 1]; // ds_load_b64, bank-conflict-free
        v2f b; b[0] = bp[0];  b[1] = bp[N];
        acc = __builtin_amdgcn_wmma_f32_16x16x4_f32(
            /*neg_a=*/false, a, /*neg_b=*/false, b,
            /*c_mod=*/(short)0, acc, /*reuse_a=*/false, /*reuse_b=*/false);
        bp += 4 * N;
    }

    float bv = 0.0f;
    if (HAS_BIAS) bv = bias[col];
    // C/D layout: VGPR v holds row M = v + 8*half, N = col
#pragma unroll
    for (int v = 0; v < 8; ++v) {
        C[(size_t)(tm * 16 + half * 8 + v) * N + col] = acc[v] + bv;
    }
}

// ---------------- aggregation ----------------

// out_i = dinv_i^2 * h_i   (self-loop term; also fully initializes out)
__global__ void selfloop_init_kernel(const float* __restrict__ h,
                                     const float* __restrict__ dinv,
                                     float* __restrict__ out, int n) {
    int i = blockIdx.x * blockDim.x + threadIdx.x;
    if (i < n) {
        float d = dinv[i >> 7];                  // node = i / 128
        out[i] = d * d * h[i];
    }
}

// one wave per edge: 32 lanes x float4 = 128 channels
__global__ void edge_scatter_kernel(const float* __restrict__ h,
                                    const long long* __restrict__ src,
                                    const long long* __restrict__ dst,
                                    const float* __restrict__ dinv,
                                    float* __restrict__ out, int e) {
    int wave = (int)((blockIdx.x * blockDim.x + threadIdx.x) >> 5);
    int lane = threadIdx.x & 31;
    if (wave >= e) return;
    int s = (int)src[wave];
    int d = (int)dst[wave];
    float norm = dinv[s] * dinv[d];
    const float4* hp = (const float4*)(h + (size_t)s * DIM) + lane;  // b128 load
    float4 v = *hp;
    float* op = out + (size_t)d * DIM + lane * 4;
    atomicAdd(op + 0, v.x * norm);
    atomicAdd(op + 1, v.y * norm);
    atomicAdd(op + 2, v.z * norm);
    atomicAdd(op + 3, v.w * norm);
}

__global__ void bias_lrelu_kernel(float* __restrict__ out,
                                  const float* __restrict__ b, int n) {
    int i = blockIdx.x * blockDim.x + threadIdx.x;
    if (i < n) {
        float v = out[i] + b[i & (DIM - 1)];
        out[i] = v > 0.0f ? v : LEAKY * v;
    }
}

// ---------------- host orchestration ----------------

extern "C" void kernel_launch(void* const* d_in, const int* in_sizes, int n_in,
                              void* d_out, int out_size, void* d_ws, size_t ws_size,
                              hipStream_t stream) {
    (void)in_sizes; (void)n_in; (void)out_size; (void)ws_size;

    const float*     x    = (const float*)d_in[0];
    const long long* ei   = (const long long*)d_in[1];     // int64 [2, E]
    const long long* srcp = ei;                            // edge_index[0]
    const long long* dstp = ei + N_EDGES;                  // edge_index[1]
    const float*     W1   = (const float*)d_in[2];
    const float*     b1   = (const float*)d_in[3];
    const float*     W2   = (const float*)d_in[4];
    const float*     b2   = (const float*)d_in[5];
    const float*     Wfc  = (const float*)d_in[6];
    const float*     bfc  = (const float*)d_in[7];
    float*           out  = (float*)d_out;

    float* bufA = (float*)d_ws;                            // [N_NODES, DIM]
    float* bufB = bufA + (size_t)N_NODES * DIM;            // [N_NODES, DIM]
    float* deg  = bufB + (size_t)N_NODES * DIM;            // [N_NODES] -> dinv in place

    const int TB = 256;
    const int nodeBlks = (N_NODES + TB - 1) / TB;
    const int edgeBlks = (N_EDGES + TB - 1) / TB;
    const int featN    = N_NODES * DIM;
    const int featBlks = (featN + TB - 1) / TB;
    const int edgeWaveBlks = (N_EDGES + (TB / 32) - 1) / (TB / 32);
    const int rowTiles = N_NODES / 16;                     // 3125 (exact)

    // normalization
    init_deg_kernel <<<nodeBlks, TB, 0, stream>>>(deg, N_NODES);
    deg_accum_kernel<<<edgeBlks, TB, 0, stream>>>(dstp, deg, N_EDGES);
    rsqrt_kernel    <<<nodeBlks, TB, 0, stream>>>(deg, N_NODES);

    // layer 1: h1 = x @ W1 ; agg ; + b1 ; lrelu
    gemm_wmma_f32<DIM, DIM, false><<<rowTiles, 256, 0, stream>>>(x, W1, nullptr, bufA);
    selfloop_init_kernel<<<featBlks, TB, 0, stream>>>(bufA, deg, bufB, featN);
    edge_scatter_kernel <<<edgeWaveBlks, TB, 0, stream>>>(bufA, srcp, dstp, deg, bufB, N_EDGES);
    bias_lrelu_kernel   <<<featBlks, TB, 0, stream>>>(bufB, b1, featN);

    // layer 2: h2 = a1 @ W2 ; agg ; + b2 ; lrelu
    gemm_wmma_f32<DIM, DIM, false><<<rowTiles, 256, 0, stream>>>(bufB, W2, nullptr, bufA);
    selfloop_init_kernel<<<featBlks, TB, 0, stream>>>(bufA, deg, bufB, featN);
    edge_scatter_kernel <<<edgeWaveBlks, TB, 0, stream>>>(bufA, srcp, dstp, deg, bufB, N_EDGES);
    bias_lrelu_kernel   <<<featBlks, TB, 0, stream>>>(bufB, b2, featN);

    // fc: out = a2 @ Wfc + bfc (4 waves per block, N=64)
    gemm_wmma_f32<DIM, FC_OUT, true><<<rowTiles, 128, 0, stream>>>(bufB, Wfc, bfc, out);
}